// HalfHadamardTrustQuantizer_82798379532944
// MI455X (gfx1250) — compile-verified
//
#include <hip/hip_runtime.h>

typedef __attribute__((ext_vector_type(2))) float v2f;
typedef __attribute__((ext_vector_type(8))) float v8f;
typedef __attribute__((ext_vector_type(4))) unsigned int u32x4;
typedef __attribute__((ext_vector_type(4))) int i32x4;
typedef __attribute__((ext_vector_type(8))) int i32x8;

#define BB 16
#define CC 512
#define HW 3136          // 56*56
#define GG 8             // CC/64
#define PT 49            // HW/64

// ---------------- kernel 1: per-batch sum of squares (Hadamard is orthogonal,
// so mean(x_had^2) == mean(x^2); no need to materialize x_had for the std) ----
__global__ void __launch_bounds__(256)
k_sumsq(const float* __restrict__ x, float* __restrict__ sumsq) {
  const int b = blockIdx.y;
  const int t = threadIdx.x;
  const float4* xb = (const float4*)(x + (size_t)b * ((size_t)CC * HW));
  const size_t base = (size_t)blockIdx.x * 2048;   // 2048 float4 per block
  float s = 0.f;
#pragma unroll
  for (int i = 0; i < 8; ++i) {
    float4 v = xb[base + (size_t)i * 256 + t];
    s += v.x * v.x + v.y * v.y + v.z * v.z + v.w * v.w;
  }
  __shared__ float red[256];
  red[t] = s;
  __syncthreads();
  for (int off = 128; off > 0; off >>= 1) {
    if (t < off) red[t] += red[t + off];
    __syncthreads();
  }
  if (t == 0) atomicAdd(&sumsq[b], red[0]);
}

// --------- Tensor Data Mover: 2D strided tile (or 1D run) -> LDS ------------
// D# per CDNA5 ISA ch.8: group0 = {count, lds_addr, global_addr, type=2},
// group1 = {data_size=4B, tensor_dim0/1, tile_dim0/1, tensor_dim0_stride},
// groups 2/3 zero (<=2D tensor). Tracked with TENSORcnt.
__device__ __attribute__((always_inline)) inline void
tdm_load_to_lds(const void* gptr, unsigned lds_off,
                unsigned tensor_d0, unsigned tensor_d1,
                unsigned tile_d0, unsigned tile_d1, unsigned stride0) {
  unsigned long long ga = (unsigned long long)gptr;
  u32x4 g0;
  g0[0] = 1u;                                          // count=1 (valid, user)
  g0[1] = lds_off;                                     // lds_addr (bytes)
  g0[2] = (unsigned)(ga & 0xFFFFFFFFu);                // global_addr[31:0]
  g0[3] = (unsigned)((ga >> 32) & 0x01FFFFFFu) | 0x80000000u; // [56:32]|type=2
  i32x8 g1;
  g1[0] = (int)(2u << 16);                             // data_size=2 (4 bytes)
  g1[1] = (int)((tensor_d0 & 0xFFFFu) << 16);          // tensor_dim0[15:0]
  g1[2] = (int)(((tensor_d0 >> 16) & 0xFFFFu) | ((tensor_d1 & 0xFFFFu) << 16));
  g1[3] = (int)(((tensor_d1 >> 16) & 0xFFFFu) | ((tile_d0 & 0xFFFFu) << 16));
  g1[4] = (int)(tile_d1 & 0xFFFFu);                    // tile_dim1; tile_dim2=0
  g1[5] = (int)stride0;                                // tensor_dim0_stride[31:0]
  g1[6] = 0;                                           // stride0[47:32], stride1 lo
  g1[7] = 0;
  i32x4 gz = {0, 0, 0, 0};
#if __clang_major__ >= 23
  i32x8 gz8 = {0, 0, 0, 0, 0, 0, 0, 0};
  __builtin_amdgcn_tensor_load_to_lds(g0, g1, gz, gz, gz8, 0);
#else
  __builtin_amdgcn_tensor_load_to_lds(g0, g1, gz, gz, 0);
#endif
}

// ---------------- kernel 2: TDM-staged WMMA Hadamard + trust quantizer ------
// One 128-thread workgroup = one (b, group, 64-pixel tile). Wave 0 issues two
// TENSOR_LOAD_TO_LDS DMAs (H 64x64, X tile 64ch x 64px with row stride 3136),
// waits TENSORcnt, then all 4 waves run V_WMMA_F32_16X16X4_F32 out of LDS.
__global__ void __launch_bounds__(128)
k_had_quant(const float* __restrict__ x, const float* __restrict__ hmat,
            const float* __restrict__ sumsq, float* __restrict__ out) {
  __shared__ float sX[64 * 64];
  __shared__ float sH[64 * 64];

  const int t  = threadIdx.x;      // 0..127
  const int b  = blockIdx.z;       // 0..15
  const int g  = blockIdx.y;       // 0..7
  const int p0 = blockIdx.x * 64;  // pixel tile base, 0..3072

  // per-batch quantizer constants (uniform -> scalar regs)
  const float ss    = sumsq[b];
  const float stdv  = sqrtf(ss * (1.0f / ((float)CC * (float)HW)));
  const float scale = 2.513930578568423f * stdv + 1e-8f;
  const float step  = scale * (2.0f / 15.0f);
  const float istep = 1.0f / step;
  const float hstep = 0.5f * step;

  const float* xg = x + ((size_t)b * CC + (size_t)g * 64) * (size_t)HW + p0;

  if (t < 32) {   // wave 0 drives the Tensor Data Mover (EXEC-independent)
    typedef __attribute__((address_space(3))) void lds_t;
    unsigned offH = (unsigned)(unsigned long long)(lds_t*)sH;
    unsigned offX = (unsigned)(unsigned long long)(lds_t*)sX;
    // H: 1D run of 4096 contiguous floats
    tdm_load_to_lds(hmat, offH, 4096u, 0u, 4096u, 0u, 4096u);
    // X: 2D tile, dim0 = 64 contiguous pixels, dim1 = 64 rows, stride 3136
    tdm_load_to_lds(xg, offX, (unsigned)HW, 64u, 64u, 64u, (unsigned)HW);
    __builtin_amdgcn_s_wait_tensorcnt(0);
  }
  __syncthreads();   // publish TDM-written LDS to the other 3 waves

  const int wave   = t >> 5;        // 0..3
  const int lane   = t & 31;
  const int half   = lane >> 4;     // hi/lo 16 lanes
  const int l16    = lane & 15;
  const int m_base = wave * 16;     // output-channel tile

  // A fragments: A[m,k] = H^T[m,k] = H[k,m].
  // ISA 16x4 f32 A layout: v0 = K (0|2), v1 = K (1|3); M = lane%16.
  v2f afr[16];
#pragma unroll
  for (int kk = 0; kk < 16; ++kk) {
    int k0 = kk * 4 + half * 2;
    afr[kk].x = sH[(k0 + 0) * 64 + (m_base + l16)];
    afr[kk].y = sH[(k0 + 1) * 64 + (m_base + l16)];
  }

  float* og = out + ((size_t)b * CC + (size_t)g * 64 + m_base) * (size_t)HW + p0;

#pragma unroll
  for (int ps = 0; ps < 4; ++ps) {          // four 16-pixel sub-tiles
    v8f acc = {};
#pragma unroll
    for (int kk = 0; kk < 16; ++kk) {       // K = 64 in steps of 4
      int k0 = kk * 4 + half * 2;
      v2f bfr;                              // B 4x16: row K, col = lane%16
      bfr.x = sX[(k0 + 0) * 64 + ps * 16 + l16];
      bfr.y = sX[(k0 + 1) * 64 + ps * 16 + l16];
      acc = __builtin_amdgcn_wmma_f32_16x16x4_f32(
          false, afr[kk], false, bfr, (short)0, acc, false, false);
    }
    // C/D layout: VGPR j -> M = j (lanes 0-15) or j+8 (lanes 16-31), N = lane%16
#pragma unroll
    for (int j = 0; j < 8; ++j) {
      int   m  = j + half * 8;
      float y  = acc[j];
      float xc = fminf(fmaxf(y, -scale), scale);
      float tq = __builtin_rintf(xc * istep + 0.5f);   // RNE like jnp.round
      float xq = tq * step - hstep;
      og[(size_t)m * HW + ps * 16 + l16] = xq;
    }
  }
}

extern "C" void kernel_launch(void* const* d_in, const int* in_sizes, int n_in,
                              void* d_out, int out_size, void* d_ws, size_t ws_size,
                              hipStream_t stream) {
  const float* x    = (const float*)d_in[0];   // (16,512,56,56) f32
  const float* hmat = (const float*)d_in[1];   // (64,64) f32
  float* out   = (float*)d_out;                // (16,512,56,56) f32
  float* sumsq = (float*)d_ws;                 // 16 floats of scratch

  hipMemsetAsync(sumsq, 0, BB * sizeof(float), stream);

  // 16 batches x 196 blocks x 256 threads x 8 float4 = full tensor
  k_sumsq<<<dim3(196, BB), 256, 0, stream>>>(x, sumsq);

  // 49 pixel-tiles x 8 groups x 16 batches, 128 threads (4 waves) each
  k_had_quant<<<dim3(PT, GG, BB), 128, 0, stream>>>(x, hmat, sumsq, out);
}